// AuditableHybridGNN_21887153340681
// MI455X (gfx1250) — compile-verified
//
#include <hip/hip_runtime.h>
#include <hip/hip_bf16.h>

#define N_ENT 100000
#define N_PSG 20000
#define N_EDGE 500000
#define D 256
#define ALPHA 0.1f
#define LN_EPS 1e-5f

typedef __attribute__((ext_vector_type(2))) float v2f;
typedef __attribute__((ext_vector_type(8))) float v8f;

// ---------------------------------------------------------------------------
// Kernel 1: h_ent = LayerNorm(0.9*h_local + 0.1*h_global); relevance = sigmoid(h_ent . q)
// One wave (32 lanes) per entity row; each lane owns 8 contiguous columns.
// ---------------------------------------------------------------------------
__global__ void __launch_bounds__(256)
ent_fuse_ln_kernel(const float* __restrict__ hl, const float* __restrict__ hg,
                   const float* __restrict__ q, const float* __restrict__ gamma,
                   const float* __restrict__ beta,
                   float* __restrict__ h_ent, float* __restrict__ rel) {
    const int wave = threadIdx.x >> 5;
    const int lane = threadIdx.x & 31;
    const long row = (long)blockIdx.x * 8 + wave;
    if (row >= N_ENT) return;

    const long base = row * D + lane * 8;
    float x[8];
    float s = 0.f;
#pragma unroll
    for (int v = 0; v < 8; ++v) {
        x[v] = (1.0f - ALPHA) * hl[base + v] + ALPHA * hg[base + v];
        s += x[v];
    }
#pragma unroll
    for (int m = 16; m >= 1; m >>= 1) s += __shfl_xor(s, m, 32);
    const float mu = s * (1.0f / (float)D);

    float vs = 0.f;
#pragma unroll
    for (int v = 0; v < 8; ++v) {
        const float d = x[v] - mu;
        vs += d * d;
    }
#pragma unroll
    for (int m = 16; m >= 1; m >>= 1) vs += __shfl_xor(vs, m, 32);
    const float rstd = rsqrtf(vs * (1.0f / (float)D) + LN_EPS);

    float dq = 0.f;
#pragma unroll
    for (int v = 0; v < 8; ++v) {
        const int c = lane * 8 + v;
        const float y = (x[v] - mu) * rstd * gamma[c] + beta[c];
        h_ent[base + v] = y;
        dq += y * q[c];
    }
#pragma unroll
    for (int m = 16; m >= 1; m >>= 1) dq += __shfl_xor(dq, m, 32);
    if (lane == 0) rel[row] = 1.0f / (1.0f + __expf(-dq));
}

// ---------------------------------------------------------------------------
// Kernel 2: zero psg_context accumulator
// ---------------------------------------------------------------------------
__global__ void zero_kernel(float* __restrict__ p, long n) {
    const long i = (long)blockIdx.x * blockDim.x + threadIdx.x;
    if (i < n) p[i] = 0.f;
}

// ---------------------------------------------------------------------------
// Kernel 3: edge scatter. One wave per edge; 8 float atomics per lane.
// psg_context is 20.5 MB -> atomics land in the 192 MB L2.
// ---------------------------------------------------------------------------
__global__ void __launch_bounds__(256)
edge_scatter_kernel(const float* __restrict__ h_ent, const float* __restrict__ rel,
                    const int* __restrict__ ent_idx, const int* __restrict__ psg_idx,
                    float* __restrict__ ctx) {
    const int wave = threadIdx.x >> 5;
    const int lane = threadIdx.x & 31;
    const long e = (long)blockIdx.x * 8 + wave;
    if (e >= N_EDGE) return;

    const int ent = ent_idx[e];
    const int psg = psg_idx[e];
    const float r = rel[ent];

    const float* __restrict__ src = h_ent + (long)ent * D + lane * 8;
    float* __restrict__ dst = ctx + (long)psg * D + lane * 8;

    const float4 a = *(const float4*)(src);
    const float4 b = *(const float4*)(src + 4);
    atomicAdd(dst + 0, a.x * r);
    atomicAdd(dst + 1, a.y * r);
    atomicAdd(dst + 2, a.z * r);
    atomicAdd(dst + 3, a.w * r);
    atomicAdd(dst + 4, b.x * r);
    atomicAdd(dst + 5, b.y * r);
    atomicAdd(dst + 6, b.z * r);
    atomicAdd(dst + 7, b.w * r);
}

// ---------------------------------------------------------------------------
// Kernel 4: h_psg = LayerNorm(h_passage + psg_context). One wave per row.
// ---------------------------------------------------------------------------
__global__ void __launch_bounds__(256)
psg_ln_kernel(const float* __restrict__ hp, const float* __restrict__ ctx,
              const float* __restrict__ gamma, const float* __restrict__ beta,
              float* __restrict__ h_psg) {
    const int wave = threadIdx.x >> 5;
    const int lane = threadIdx.x & 31;
    const long row = (long)blockIdx.x * 8 + wave;
    if (row >= N_PSG) return;

    const long base = row * D + lane * 8;
    float x[8];
    float s = 0.f;
#pragma unroll
    for (int v = 0; v < 8; ++v) {
        x[v] = hp[base + v] + ctx[base + v];
        s += x[v];
    }
#pragma unroll
    for (int m = 16; m >= 1; m >>= 1) s += __shfl_xor(s, m, 32);
    const float mu = s * (1.0f / (float)D);

    float vs = 0.f;
#pragma unroll
    for (int v = 0; v < 8; ++v) {
        const float d = x[v] - mu;
        vs += d * d;
    }
#pragma unroll
    for (int m = 16; m >= 1; m >>= 1) vs += __shfl_xor(vs, m, 32);
    const float rstd = rsqrtf(vs * (1.0f / (float)D) + LN_EPS);

#pragma unroll
    for (int v = 0; v < 8; ++v) {
        const int c = lane * 8 + v;
        h_psg[base + v] = (x[v] - mu) * rstd * gamma[c] + beta[c];
    }
}

// ---------------------------------------------------------------------------
// Kernel 5: cadd[j] = b1[j] + sum_k q[k] * W1[256+k][j]  (passage-invariant term)
// ---------------------------------------------------------------------------
__global__ void cadd_kernel(const float* __restrict__ q, const float* __restrict__ W1,
                            const float* __restrict__ b1, float* __restrict__ cadd) {
    const int j = threadIdx.x;
    float s = b1[j];
    for (int k = 0; k < D; ++k) s += q[k] * W1[(long)(D + k) * D + j];
    cadd[j] = s;
}

// ---------------------------------------------------------------------------
// Kernel 6: WMMA scoring head.
//   hid = relu(h_psg @ W1[0:256] + cadd);  scores = hid @ W2 + b2
// Block = 512 threads = 16 waves; wave w owns output columns [16w,16w+16) of a
// 16-row tile. K loop: 64 x V_WMMA_F32_16X16X4_F32 (exact fp32 math).
// A layout (16x4 f32): lane holds A[lane%16][2*(lane/16)+v], v=0..1.
// B layout (4x16 f32): lane holds B[2*(lane/16)+v][lane%16].
// C/D layout: VGPR v, lane l -> row v + 8*(l/16), col l%16.
// ---------------------------------------------------------------------------
__global__ void __launch_bounds__(512)
score_wmma_kernel(const float* __restrict__ h_psg, const float* __restrict__ W1,
                  const float* __restrict__ W2, const float* __restrict__ b2,
                  const float* __restrict__ cadd, float* __restrict__ scores) {
    __shared__ float srow[16];
    const int lane = threadIdx.x & 31;
    const int wave = threadIdx.x >> 5;   // 0..15 = column tile
    const int n = lane & 15;             // B col / A row within tile
    const int h = lane >> 4;             // half-wave select
    const int R = blockIdx.x * 16;       // row tile base (N_PSG % 16 == 0)
    const int colBase = wave * 16;

    if (threadIdx.x < 16) srow[threadIdx.x] = 0.f;
    __syncthreads();

    const float* __restrict__ Arow = h_psg + (long)(R + n) * D;

    v8f acc = {};
    for (int k0 = 0; k0 < D; k0 += 4) {
        const int ka = k0 + 2 * h;
        v2f a = *(const v2f*)(Arow + ka);                 // A[m][ka], A[m][ka+1]
        v2f b;
        b.x = W1[(long)ka * D + colBase + n];             // B[ka][n]
        b.y = W1[(long)(ka + 1) * D + colBase + n];       // B[ka+1][n]
        acc = __builtin_amdgcn_wmma_f32_16x16x4_f32(
            /*neg_a=*/false, a, /*neg_b=*/false, b,
            /*c_mod=*/(short)0, acc, /*reuse_a=*/false, /*reuse_b=*/false);
    }

    const float w2 = W2[colBase + n];
    const float ca = cadd[colBase + n];
    float p[8];
#pragma unroll
    for (int v = 0; v < 8; ++v) {
        float val = acc[v] + ca;
        val = val > 0.f ? val : 0.f;     // ReLU
        p[v] = val * w2;
    }
    // Reduce across the 16 lanes sharing the same half (cols of this tile).
#pragma unroll
    for (int m = 1; m <= 8; m <<= 1) {
#pragma unroll
        for (int v = 0; v < 8; ++v) p[v] += __shfl_xor(p[v], m, 32);
    }
    if (n == 0) {
#pragma unroll
        for (int v = 0; v < 8; ++v) atomicAdd(&srow[v + 8 * h], p[v]);
    }
    __syncthreads();
    if (threadIdx.x < 16) scores[R + threadIdx.x] = srow[threadIdx.x] + b2[0];
}

// ---------------------------------------------------------------------------
// Launch
// ---------------------------------------------------------------------------
extern "C" void kernel_launch(void* const* d_in, const int* in_sizes, int n_in,
                              void* d_out, int out_size, void* d_ws, size_t ws_size,
                              hipStream_t stream) {
    const float* h_local  = (const float*)d_in[0];
    const float* h_global = (const float*)d_in[1];
    const float* h_pass   = (const float*)d_in[2];
    const float* q        = (const float*)d_in[3];
    const int*   ent_idx  = (const int*)d_in[4];
    const int*   psg_idx  = (const int*)d_in[5];
    const float* gamma_e  = (const float*)d_in[6];
    const float* beta_e   = (const float*)d_in[7];
    const float* gamma_p  = (const float*)d_in[8];
    const float* beta_p   = (const float*)d_in[9];
    const float* W1       = (const float*)d_in[10];
    const float* b1       = (const float*)d_in[11];
    const float* W2       = (const float*)d_in[12];
    const float* b2       = (const float*)d_in[13];
    float* scores = (float*)d_out;

    // Workspace layout (fp32): h_ent | relevance | psg_context | h_psg | cadd
    float* ws     = (float*)d_ws;
    float* h_ent  = ws;                               // N_ENT * D
    float* rel    = h_ent + (size_t)N_ENT * D;        // N_ENT
    float* ctx    = rel + N_ENT;                      // N_PSG * D
    float* h_psg  = ctx + (size_t)N_PSG * D;          // N_PSG * D
    float* cadd   = h_psg + (size_t)N_PSG * D;        // D

    // 1) entity blend + LN + relevance gate (one wave per row)
    ent_fuse_ln_kernel<<<(N_ENT + 7) / 8, 256, 0, stream>>>(
        h_local, h_global, q, gamma_e, beta_e, h_ent, rel);

    // 2) zero the scatter accumulator
    {
        const long n = (long)N_PSG * D;
        zero_kernel<<<(unsigned)((n + 255) / 256), 256, 0, stream>>>(ctx, n);
    }

    // 3) edge gather/scale/scatter-add (one wave per edge, L2-resident atomics)
    edge_scatter_kernel<<<(N_EDGE + 7) / 8, 256, 0, stream>>>(
        h_ent, rel, ent_idx, psg_idx, ctx);

    // 4) passage LN
    psg_ln_kernel<<<(N_PSG + 7) / 8, 256, 0, stream>>>(
        h_pass, ctx, gamma_p, beta_p, h_psg);

    // 5) passage-invariant MLP term: q @ W1[256:512] + b1
    cadd_kernel<<<1, D, 0, stream>>>(q, W1, b1, cadd);

    // 6) WMMA fp32 scoring head
    score_wmma_kernel<<<N_PSG / 16, 512, 0, stream>>>(
        h_psg, W1, W2, b2, cadd, scores);
}